// GADGNN_6803228197649
// MI455X (gfx1250) — compile-verified
//
#include <hip/hip_runtime.h>
#include <hip/hip_bf16.h>

typedef float v2f __attribute__((ext_vector_type(2)));
typedef float v8f __attribute__((ext_vector_type(8)));

#define LRELU_SLOPE 0.01f

// ---------------------------------------------------------------------------
// Generic WMMA f32 GEMM: C = epilogue(A @ B + bias)
//   A given as up to 4 "parts", each holding 64 K-columns with row stride lda.
//   B is [Kdim, Ncols] row-major, Ncols % 64 == 0.
//   One wave computes a 16-row x 64-col strip = 4 x V_WMMA_F32_16X16X4_F32
//   tiles, so each A fragment load feeds 4 WMMA issues (4x less A traffic).
// ---------------------------------------------------------------------------
__global__ void wmma_gemm_f32(const float* __restrict__ A0,
                              const float* __restrict__ A1,
                              const float* __restrict__ A2,
                              const float* __restrict__ A3,
                              int lda, int Kdim,
                              const float* __restrict__ B, int ldb,
                              const float* __restrict__ bias,
                              const float* __restrict__ residual, int ldres,
                              float* __restrict__ C, int ldo,
                              int Mrows, int Ncols, int act)
{
    const int lane = threadIdx.x & 31;
    const int gw   = (blockIdx.x * blockDim.x + threadIdx.x) >> 5;
    const int nw   = (gridDim.x * blockDim.x) >> 5;
    const int colStrips = Ncols >> 6;               // strips of 64 columns
    const int rowTiles  = (Mrows + 15) >> 4;
    const int strips = colStrips * rowTiles;
    const bool hi  = lane >= 16;
    const int koff = hi ? 2 : 0;
    const int ln16 = lane & 15;

    for (int strip = gw; strip < strips; strip += nw) {
        const int rt = strip / colStrips;
        const int cs = strip - rt * colStrips;
        const int row0 = rt << 4;
        const int col0 = cs << 6;
        int m = row0 + ln16;
        int mc = (m < Mrows) ? m : (Mrows - 1);     // clamp; masked on store

        v8f acc[4];
#pragma unroll
        for (int ct = 0; ct < 4; ++ct)
            acc[ct] = (v8f){0.f, 0.f, 0.f, 0.f, 0.f, 0.f, 0.f, 0.f};

        for (int part = 0; part * 64 < Kdim; ++part) {
            const float* Ap = (part == 0) ? A0 : (part == 1) ? A1
                            : (part == 2) ? A2 : A3;
            const float* arow = Ap + (size_t)mc * lda;
            const int kbase = part * 64;
            const int kend = (Kdim - kbase < 64) ? (Kdim - kbase) : 64;
            for (int kl = 0; kl < kend; kl += 4) {
                v2f a;
                a.x = arow[kl + koff];
                a.y = arow[kl + koff + 1];
                const float* brow = B + (size_t)(kbase + kl + koff) * ldb + col0 + ln16;
#pragma unroll
                for (int ct = 0; ct < 4; ++ct) {
                    v2f b;
                    b.x = brow[ct * 16];
                    b.y = brow[ct * 16 + ldb];
                    acc[ct] = __builtin_amdgcn_wmma_f32_16x16x4_f32(
                                false, a, false, b, (short)0, acc[ct], false, false);
                }
            }
        }

#pragma unroll
        for (int ct = 0; ct < 4; ++ct) {
            const int n = col0 + ct * 16 + ln16;
            const float bv = bias ? bias[n] : 0.0f;
#pragma unroll
            for (int r = 0; r < 8; ++r) {
                int rr = row0 + r + (hi ? 8 : 0);
                if (rr < Mrows) {
                    float v = acc[ct][r] + bv;
                    if (act) v = (v > 0.0f) ? v : LRELU_SLOPE * v;
                    if (residual) v += residual[(size_t)rr * ldres + n];
                    C[(size_t)rr * ldo + n] = v;
                }
            }
        }
    }
}

// ---------------------------------------------------------------------------
// Small utility kernels
// ---------------------------------------------------------------------------
__global__ void fill_kernel(float* p, long n, float v) {
    long i = (long)blockIdx.x * blockDim.x + threadIdx.x;
    long stride = (long)gridDim.x * blockDim.x;
    for (; i < n; i += stride) p[i] = v;
}

__global__ void neg_copy_kernel(const float* __restrict__ in, float* __restrict__ out, long n) {
    long i = (long)blockIdx.x * blockDim.x + threadIdx.x;
    long stride = (long)gridDim.x * blockDim.x;
    for (; i < n; i += stride) out[i] = -in[i];
}

__global__ void deg_kernel(const int* __restrict__ dst, float* __restrict__ deg, int E) {
    int i = blockIdx.x * blockDim.x + threadIdx.x;
    int stride = gridDim.x * blockDim.x;
    for (; i < E; i += stride) atomicAdd(&deg[dst[i]], 1.0f);
}

__global__ void dinv_kernel(const float* __restrict__ deg, float* __restrict__ dinv, int N) {
    int i = blockIdx.x * blockDim.x + threadIdx.x;
    int stride = gridDim.x * blockDim.x;
    for (; i < N; i += stride) {
        float d = deg[i];
        dinv[i] = (d > 0.0f) ? rsqrtf(fmaxf(d, 1.0f)) : 0.0f;
    }
}

__global__ void norm_kernel(const int* __restrict__ src, const int* __restrict__ dst,
                            const float* __restrict__ dinv, float* __restrict__ norm, int E) {
    int i = blockIdx.x * blockDim.x + threadIdx.x;
    int stride = gridDim.x * blockDim.x;
    for (; i < E; i += stride) norm[i] = -(dinv[src[i]] * dinv[dst[i]]);
}

// out[dst] += scale * norm[e] * x[src]  over F=64 columns (4 cols per thread)
__global__ void edge_scatter_kernel(const int* __restrict__ src, const int* __restrict__ dst,
                                    const float* __restrict__ norm,
                                    const float* __restrict__ X, float* __restrict__ Out,
                                    int E, float scale) {
    long total = (long)E * 16;
    long i = (long)blockIdx.x * blockDim.x + threadIdx.x;
    long stride = (long)gridDim.x * blockDim.x;
    for (; i < total; i += stride) {
        int e  = (int)(i >> 4);
        int cg = (int)(i & 15) << 2;
        int s = src[e], d = dst[e];
        float w = norm[e] * scale;
        const float4 x = *(const float4*)(X + (size_t)s * 64 + cg);
        float* o = Out + (size_t)d * 64 + cg;
        atomicAdd(o + 0, w * x.x);
        atomicAdd(o + 1, w * x.y);
        atomicAdd(o + 2, w * x.z);
        atomicAdd(o + 3, w * x.w);
    }
}

// Build Wcat [192,256] and bcat [256] from cheb_w [4,3,64,64], cheb_b [4,64]
__global__ void pack_wcat_kernel(const float* __restrict__ cheb_w,
                                 const float* __restrict__ cheb_b,
                                 float* __restrict__ Wcat, float* __restrict__ bcat) {
    int idx = blockIdx.x * blockDim.x + threadIdx.x;
    if (idx < 192 * 256) {
        int kk = idx >> 8;           // 0..191
        int jj = idx & 255;          // 0..255
        int t = kk >> 6, k = kk & 63;
        int i = jj >> 6, c = jj & 63;
        Wcat[idx] = cheb_w[(((size_t)(i * 3 + t) * 64) + k) * 64 + c];
    }
    if (idx < 256) {
        int i = idx >> 6, c = idx & 63;
        bcat[idx] = cheb_b[i * 64 + c];
    }
}

// scores[n] = dot(h4[n,0:128], tmp[batch[n],0:128]) ; one wave per node
__global__ void scores_kernel(const float* __restrict__ h4, const float* __restrict__ tmp,
                              const int* __restrict__ batch, float* __restrict__ scores, int N) {
    int lane = threadIdx.x & 31;
    int gw = (blockIdx.x * blockDim.x + threadIdx.x) >> 5;
    int nw = (gridDim.x * blockDim.x) >> 5;
    for (int node = gw; node < N; node += nw) {
        int b = batch[node];
        float s = 0.0f;
        const float* hp = h4 + (size_t)node * 128;
        const float* tp = tmp + (size_t)b * 128;
        for (int c = lane; c < 128; c += 32) s += hp[c] * tp[c];
        for (int off = 16; off > 0; off >>= 1) s += __shfl_down(s, off, 32);
        if (lane == 0) scores[node] = s;
    }
}

__global__ void counts_kernel(const int* __restrict__ batch, float* __restrict__ counts, int N) {
    int i = blockIdx.x * blockDim.x + threadIdx.x;
    int stride = gridDim.x * blockDim.x;
    for (; i < N; i += stride) atomicAdd(&counts[batch[i]], 1.0f);
}

// hb[batch[n], c] += (scores[n]/max(counts,1)) * h4[n,c]   (c in 0..127, hb ld = 256)
__global__ void pool_kernel(const float* __restrict__ h4, const float* __restrict__ scores,
                            const float* __restrict__ counts, const int* __restrict__ batch,
                            float* __restrict__ hb, int N) {
    long total = (long)N * 128;
    long i = (long)blockIdx.x * blockDim.x + threadIdx.x;
    long stride = (long)gridDim.x * blockDim.x;
    for (; i < total; i += stride) {
        int node = (int)(i >> 7);
        int c = (int)(i & 127);
        int b = batch[node];
        float w = scores[node] / fmaxf(counts[b], 1.0f);
        atomicAdd(&hb[(size_t)b * 256 + c], w * h4[(size_t)node * 128 + c]);
    }
}

// BatchNorm1d (training stats, biased var) over rows; one block per column
__global__ void bn_kernel(float* __restrict__ hb, const float* __restrict__ gamma,
                          const float* __restrict__ beta, int Brows, int Ccols) {
    __shared__ float s_sum[256];
    __shared__ float s_sq[256];
    int j = blockIdx.x;
    float sum = 0.0f, sq = 0.0f;
    for (int r = threadIdx.x; r < Brows; r += blockDim.x) {
        float v = hb[(size_t)r * Ccols + j];
        sum += v; sq += v * v;
    }
    s_sum[threadIdx.x] = sum; s_sq[threadIdx.x] = sq;
    __syncthreads();
    for (int off = blockDim.x >> 1; off > 0; off >>= 1) {
        if ((int)threadIdx.x < off) {
            s_sum[threadIdx.x] += s_sum[threadIdx.x + off];
            s_sq[threadIdx.x]  += s_sq[threadIdx.x + off];
        }
        __syncthreads();
    }
    float mu = s_sum[0] / (float)Brows;
    float var = s_sq[0] / (float)Brows - mu * mu;
    float inv = rsqrtf(var + 1e-5f);
    float g = gamma[j], be = beta[j];
    for (int r = threadIdx.x; r < Brows; r += blockDim.x) {
        float v = hb[(size_t)r * Ccols + j];
        hb[(size_t)r * Ccols + j] = (v - mu) * inv * g + be;
    }
}

// out[b, c] = hb[b,:] @ W7[:,c] + b7[c]   ([B,256] x [256,2])
__global__ void final_kernel(const float* __restrict__ hb, const float* __restrict__ W7,
                             const float* __restrict__ b7, float* __restrict__ out,
                             int Brows, int Kdim, int Ncls) {
    int i = blockIdx.x * blockDim.x + threadIdx.x;
    int total = Brows * Ncls;
    if (i >= total) return;
    int b = i / Ncls, c = i - b * Ncls;
    float s = b7[c];
    const float* hp = hb + (size_t)b * Kdim;
    for (int k = 0; k < Kdim; ++k) s += hp[k] * W7[(size_t)k * Ncls + c];
    out[i] = s;
}

// ---------------------------------------------------------------------------
// Launch
// ---------------------------------------------------------------------------
static inline int cdiv(long a, long b) { return (int)((a + b - 1) / b); }

static void launch_gemm(const float* A0, const float* A1, const float* A2, const float* A3,
                        int lda, int Kdim, const float* B, int ldb,
                        const float* bias, const float* residual, int ldres,
                        float* C, int ldo, int M, int Ncols, int act, hipStream_t stream) {
    long strips = (long)((M + 15) / 16) * (Ncols / 64);
    int blocks = cdiv(strips, 8);        // 8 waves (256 threads) per block
    if (blocks < 1) blocks = 1;
    if (blocks > 65535) blocks = 65535;
    wmma_gemm_f32<<<blocks, 256, 0, stream>>>(A0, A1, A2, A3, lda, Kdim, B, ldb,
                                              bias, residual, ldres, C, ldo, M, Ncols, act);
}

extern "C" void kernel_launch(void* const* d_in, const int* in_sizes, int n_in,
                              void* d_out, int out_size, void* d_ws, size_t ws_size,
                              hipStream_t stream) {
    const int F = 64, H = 128, NCLASS = 2;
    const int N = in_sizes[0] / F;        // 50000
    const int Bg = in_sizes[1] / F;       // 512
    const int E = in_sizes[2] / 2;        // 800000

    const float* features = (const float*)d_in[0];
    const float* xLx      = (const float*)d_in[1];
    const int*   eidx     = (const int*)d_in[2];
    const int*   batch    = (const int*)d_in[3];
    const float* W1 = (const float*)d_in[4];  const float* b1 = (const float*)d_in[5];
    const float* W2 = (const float*)d_in[6];  const float* b2 = (const float*)d_in[7];
    const float* cheb_w = (const float*)d_in[8];
    const float* cheb_b = (const float*)d_in[9];
    const float* W3 = (const float*)d_in[10]; const float* b3 = (const float*)d_in[11];
    const float* W4 = (const float*)d_in[12]; const float* b4 = (const float*)d_in[13];
    const float* W5 = (const float*)d_in[14]; const float* b5 = (const float*)d_in[15];
    const float* W6 = (const float*)d_in[16]; const float* b6 = (const float*)d_in[17];
    const float* W7 = (const float*)d_in[18]; const float* b7 = (const float*)d_in[19];
    const float* W8 = (const float*)d_in[20]; const float* b8 = (const float*)d_in[21];
    const float* W9 = (const float*)d_in[22]; const float* b9 = (const float*)d_in[23];
    const float* bn_g = (const float*)d_in[24];
    const float* bn_b = (const float*)d_in[25];

    const int* src = eidx;
    const int* dst = eidx + E;

    // workspace carve (floats)
    float* ws = (float*)d_ws;
    size_t off = 0;
    float* h1   = ws + off; off += (size_t)N * 64;
    float* h    = ws + off; off += (size_t)N * 64;     // Tx0
    float* Tx1  = ws + off; off += (size_t)N * 64;
    float* Tx2  = ws + off; off += (size_t)N * 64;
    float* hcat = ws + off; off += (size_t)N * 256;
    float* h3   = ws + off; off += (size_t)N * 128;
    float* h4   = ws + off; off += (size_t)N * 128;
    float* deg    = ws + off; off += N;
    float* dinv   = ws + off; off += N;
    float* scores = ws + off; off += N;
    float* norm   = ws + off; off += E;
    float* Wcat = ws + off; off += 192 * 256;
    float* bcat = ws + off; off += 256;
    float* tmp1 = ws + off; off += (size_t)Bg * 128;
    float* tmp2 = ws + off; off += (size_t)Bg * 128;
    float* xlx1 = ws + off; off += (size_t)Bg * 128;
    float* hb   = ws + off; off += (size_t)Bg * 256;
    float* counts = ws + off; off += Bg;
    (void)ws_size; (void)n_in; (void)out_size;

    // ---- degree / normalization coefficients ----
    fill_kernel<<<256, 256, 0, stream>>>(deg, N, 0.0f);
    deg_kernel<<<cdiv(E, 256), 256, 0, stream>>>(dst, deg, E);
    dinv_kernel<<<cdiv(N, 256), 256, 0, stream>>>(deg, dinv, N);
    norm_kernel<<<cdiv(E, 256), 256, 0, stream>>>(src, dst, dinv, norm, E);

    // ---- h = act(x@W1+b1); h = act(h@W2+b2) + residual ----
    launch_gemm(features, features, features, features, 64, 64, W1, 64,
                b1, nullptr, 0, h1, 64, N, 64, 1, stream);
    launch_gemm(h1, h1, h1, h1, 64, 64, W2, 64,
                b2, h1, 64, h, 64, N, 64, 1, stream);

    // ---- Chebyshev recursion (Tx1 = L@h ; Tx2 = 2 L@Tx1 - h) ----
    fill_kernel<<<1024, 256, 0, stream>>>(Tx1, (long)N * 64, 0.0f);
    edge_scatter_kernel<<<4096, 256, 0, stream>>>(src, dst, norm, h, Tx1, E, 1.0f);
    neg_copy_kernel<<<1024, 256, 0, stream>>>(h, Tx2, (long)N * 64);
    edge_scatter_kernel<<<4096, 256, 0, stream>>>(src, dst, norm, Tx1, Tx2, E, 2.0f);

    // ---- hcat = [h|Tx1|Tx2] @ Wcat + bcat  (all 4 heads x 3 orders fused) ----
    pack_wcat_kernel<<<cdiv(192 * 256, 256), 256, 0, stream>>>(cheb_w, cheb_b, Wcat, bcat);
    launch_gemm(h, Tx1, Tx2, Tx2, 64, 192, Wcat, 256,
                bcat, nullptr, 0, hcat, 256, N, 256, 0, stream);

    // ---- h3 = act(hcat@W3+b3) ; h4 = act(h3@W4+b4) ----
    launch_gemm(hcat, hcat + 64, hcat + 128, hcat + 192, 256, 256, W3, 128,
                b3, nullptr, 0, h3, 128, N, 128, 1, stream);
    launch_gemm(h3, h3 + 64, h3, h3, 128, 128, W4, 128,
                b4, nullptr, 0, h4, 128, N, 128, 1, stream);

    // ---- attention vectors: tmp = act(act(xLx@W8+b8)@W9+b9) ----
    launch_gemm(xLx, xLx, xLx, xLx, 64, 64, W8, 128,
                b8, nullptr, 0, tmp1, 128, Bg, 128, 1, stream);
    launch_gemm(tmp1, tmp1 + 64, tmp1, tmp1, 128, 128, W9, 128,
                b9, nullptr, 0, tmp2, 128, Bg, 128, 1, stream);

    // ---- scores, counts, weighted pooling into hb[:, 0:128] ----
    scores_kernel<<<1024, 256, 0, stream>>>(h4, tmp2, batch, scores, N);
    fill_kernel<<<4, 256, 0, stream>>>(counts, Bg, 0.0f);
    counts_kernel<<<cdiv(N, 256), 256, 0, stream>>>(batch, counts, N);
    fill_kernel<<<256, 256, 0, stream>>>(hb, (long)Bg * 256, 0.0f);

    // ---- xlx path into hb[:, 128:256] ----
    launch_gemm(xLx, xLx, xLx, xLx, 64, 64, W5, 128,
                b5, nullptr, 0, xlx1, 128, Bg, 128, 0, stream);
    launch_gemm(xlx1, xlx1 + 64, xlx1, xlx1, 128, 128, W6, 128,
                b6, nullptr, 0, hb + 128, 256, Bg, 128, 1, stream);

    pool_kernel<<<4096, 256, 0, stream>>>(h4, scores, counts, batch, hb, N);

    // ---- BatchNorm + classifier ----
    bn_kernel<<<2 * H, 256, 0, stream>>>(hb, bn_g, bn_b, Bg, 2 * H);
    final_kernel<<<cdiv((long)Bg * NCLASS, 256), 256, 0, stream>>>(
        hb, W7, b7, (float*)d_out, Bg, 2 * H, NCLASS);
}